// MPSLayer_47167330844706
// MI455X (gfx1250) — compile-verified
//
#include <hip/hip_runtime.h>

typedef __attribute__((ext_vector_type(2))) float v2f;
typedef __attribute__((ext_vector_type(8))) float v8f;

// D = A(16x4 f32) * B(4x16 f32) + C(16x16 f32), full fp32 precision.
// (neg_a, A, neg_b, B, c_mod, C, reuse_a, reuse_b)
#define WMMA_F32X4(Areg, Breg, Creg)                                           \
  __builtin_amdgcn_wmma_f32_16x16x4_f32(false, (Areg), false, (Breg),          \
                                        (short)0, (Creg), false, false)

// ds_swizzle group-of-32: new_lane = (lane & 0x1f) ^ 0x10  (swap lane halves)
#define SWAPX16(f)                                                             \
  __int_as_float(__builtin_amdgcn_ds_swizzle(__float_as_int(f), 0x401f))

// State layout: v8f element kk (kk=0..3) holds v_state[e=2kk, b] in lanes 0-15
// and v_state[e=2kk+1, b] in lanes 16-31. Elements 4..7 are redundant dups.
// This is the natural WMMA D output when the A (weight) matrix row M carries
// bond index e(M) = 2*(M&3) + (M>>3)  (rows 4-7 / 12-15 duplicate 0-3 / 8-11),
// and it is directly the B operand of the next site: B chunk kk, VGPR j, lane
// half hi maps to k = 4kk+j+2hi -> (d = 2kk+hi, f = j), i.e. st[kk]*phi[j].

__global__ __launch_bounds__(32) void mps_chain_kernel(
    const float* __restrict__ x,        // [B, N]  (N=784)
    const float* __restrict__ w0,       // [2, 8]
    const float* __restrict__ W_left,   // [391, 8, 2, 8]
    const float* __restrict__ w_label,  // [8, 2, 8, 10]
    const float* __restrict__ W_right,  // [390, 8, 2, 8]
    const float* __restrict__ w_last,   // [8, 2]
    float* __restrict__ out)            // [B, 10]
{
  constexpr int N = 784;
  constexpr int LABEL = 392;   // N/2
  constexpr int O = 10;

  const int lane = threadIdx.x & 31;
  const int bsub = lane & 15;                    // batch column (WMMA N dim)
  const int hi   = lane >> 4;                    // lane half -> K+2 / M+8
  const int eA   = 2 * (lane & 3) + ((lane >> 3) & 1);  // A-row bond index e(M)

  const int b = blockIdx.x * 16 + bsub;
  const float* __restrict__ xb = x + (size_t)b * N;

  // ---------------- site 0: st[kk] = phi . w0[:, 2kk+hi] --------------------
  float p = xb[0];
  v8f st;
#pragma unroll
  for (int kk = 0; kk < 4; ++kk) {
    const int d = 2 * kk + hi;
    const float val = (1.0f - p) * w0[d] + p * w0[8 + d];
    st[kk] = val;
    st[kk + 4] = val;   // keep dup halves consistent (unused)
  }

  // ---------------- left sweep: sites 1 .. LABEL-1 (391 tensors) ------------
  for (int i = 1; i < LABEL; ++i) {
    p = xb[i];
    const float ph0 = 1.0f - p, ph1 = p;
    const float* __restrict__ wb = W_left + (size_t)(i - 1) * 128;

    v2f A[4];
#pragma unroll
    for (int kk = 0; kk < 4; ++kk) {
      const int d = 2 * kk + hi;
      A[kk].x = wb[d * 16 + 0 + eA];   // f = 0
      A[kk].y = wb[d * 16 + 8 + eA];   // f = 1
    }

    v8f acc = {0.f, 0.f, 0.f, 0.f, 0.f, 0.f, 0.f, 0.f};
#pragma unroll
    for (int kk = 0; kk < 4; ++kk) {
      v2f Bm;
      Bm.x = st[kk] * ph0;
      Bm.y = st[kk] * ph1;
      acc = WMMA_F32X4(A[kk], Bm, acc);
    }
    st = acc;
  }

  // ---------------- label site: widen state to O columns --------------------
  p = xb[LABEL];
  v8f sto[O];
  {
    const float ph0 = 1.0f - p, ph1 = p;
    v2f Bm[4];
#pragma unroll
    for (int kk = 0; kk < 4; ++kk) {
      Bm[kk].x = st[kk] * ph0;
      Bm[kk].y = st[kk] * ph1;
    }
#pragma unroll
    for (int o = 0; o < O; ++o) {
      v8f acc = {0.f, 0.f, 0.f, 0.f, 0.f, 0.f, 0.f, 0.f};
#pragma unroll
      for (int kk = 0; kk < 4; ++kk) {
        const int d = 2 * kk + hi;
        v2f A;
        A.x = w_label[d * 160 + 0  + eA * 10 + o];   // f=0
        A.y = w_label[d * 160 + 80 + eA * 10 + o];   // f=1
        acc = WMMA_F32X4(A, Bm[kk], acc);
      }
      sto[o] = acc;
    }
  }

  // ---------------- right sweep: sites LABEL+1 .. N-2 (390 tensors) ---------
  for (int i = LABEL + 1; i < N - 1; ++i) {
    p = xb[i];
    const float ph0 = 1.0f - p, ph1 = p;
    const float* __restrict__ wb = W_right + (size_t)(i - (LABEL + 1)) * 128;

    v2f A[4];
#pragma unroll
    for (int kk = 0; kk < 4; ++kk) {
      const int d = 2 * kk + hi;
      A[kk].x = wb[d * 16 + 0 + eA];
      A[kk].y = wb[d * 16 + 8 + eA];
    }

#pragma unroll
    for (int o = 0; o < O; ++o) {       // 10 independent chains -> ILP
      v8f acc = {0.f, 0.f, 0.f, 0.f, 0.f, 0.f, 0.f, 0.f};
#pragma unroll
      for (int kk = 0; kk < 4; ++kk) {
        v2f Bm;
        Bm.x = sto[o][kk] * ph0;
        Bm.y = sto[o][kk] * ph1;
        acc = WMMA_F32X4(A[kk], Bm, acc);
      }
      sto[o] = acc;
    }
  }

  // ------- last site: rr[kk] = phi . w_last[2kk+hi,:]; reduce halves --------
  p = xb[N - 1];
  float rr[4];
#pragma unroll
  for (int kk = 0; kk < 4; ++kk) {
    const int d = 2 * kk + hi;
    rr[kk] = (1.0f - p) * w_last[2 * d] + p * w_last[2 * d + 1];
  }

#pragma unroll
  for (int o = 0; o < O; ++o) {
    // low lanes sum even-d terms, high lanes odd-d terms; combine via swizzle
    float s = sto[o][0] * rr[0];
    s = fmaf(sto[o][1], rr[1], s);
    s = fmaf(sto[o][2], rr[2], s);
    s = fmaf(sto[o][3], rr[3], s);
    s += SWAPX16(s);                    // full-wave exec here
    if (lane < 16)
      out[(size_t)b * O + o] = s;
  }
}

extern "C" void kernel_launch(void* const* d_in, const int* in_sizes, int n_in,
                              void* d_out, int out_size, void* d_ws, size_t ws_size,
                              hipStream_t stream) {
  const float* x       = (const float*)d_in[0];
  const float* w0      = (const float*)d_in[1];
  const float* W_left  = (const float*)d_in[2];
  const float* w_label = (const float*)d_in[3];
  const float* W_right = (const float*)d_in[4];
  const float* w_last  = (const float*)d_in[5];
  float* out = (float*)d_out;

  const int N = 784;
  const int B = in_sizes[0] / N;       // 16384
  const int blocks = B / 16;           // one wave per 16 batch rows

  mps_chain_kernel<<<blocks, 32, 0, stream>>>(x, w0, W_left, w_label,
                                              W_right, w_last, out);
}